// LoraLinear_14139032338753
// MI455X (gfx1250) — compile-verified
//
#include <hip/hip_runtime.h>

// LoRA multi-adapter fused update for MI455X (gfx1250, wave32).
// Pipeline: [init] -> [count] -> [prefix] -> [scatter] -> [shrink WMMA] -> [expand WMMA]
// Exact fp32 math via V_WMMA_F32_16X16X4_F32. Memory-bound (~406 MB HBM, ~18us floor).

typedef float v2f __attribute__((ext_vector_type(2)));
typedef float v8f __attribute__((ext_vector_type(8)));

#define T_TOK 8192
#define H_DIM 4096
#define R_RANK 16
#define O_DIM 4096
#define N_AD 8

// ---------------- workspace layout (bytes) ----------------
// 0    : counts  int[8]
// 32   : offsets int[8]
// 64   : cursor  int[8]
// 128  : perm    int[T_TOK]                (32768 B)
// 33024: vbuf    float[(T_TOK+16)*R_RANK]  (525312 B)  -- padded for tile spill reads
#define WS_OFF_COUNTS  0
#define WS_OFF_OFFSETS 32
#define WS_OFF_CURSOR  64
#define WS_OFF_PERM    128
#define WS_OFF_VBUF    33024

__global__ __launch_bounds__(32)
void lora_init(int* counts, int* cursor) {
    int i = threadIdx.x;
    if (i < N_AD) { counts[i] = 0; cursor[i] = 0; }
}

__global__ __launch_bounds__(256)
void lora_count(const int* __restrict__ idx, int* __restrict__ counts) {
    int t = blockIdx.x * 256 + threadIdx.x;
    if (t < T_TOK) atomicAdd(&counts[idx[t]], 1);
}

__global__ __launch_bounds__(32)
void lora_prefix(const int* __restrict__ counts, int* __restrict__ offsets) {
    if (threadIdx.x == 0) {
        int s = 0;
        for (int i = 0; i < N_AD; ++i) { offsets[i] = s; s += counts[i]; }
    }
}

__global__ __launch_bounds__(256)
void lora_scatter(const int* __restrict__ idx, const int* __restrict__ offsets,
                  int* __restrict__ cursor, int* __restrict__ perm) {
    int t = blockIdx.x * 256 + threadIdx.x;
    if (t >= T_TOK) return;
    int a = idx[t];
    int p = atomicAdd(&cursor[a], 1);
    perm[offsets[a] + p] = t;   // order within a segment is irrelevant (per-token math)
}

// ---- shrink: V[pos, 0..15] = x[perm[pos]] @ lora_a[a]   (one wave per 16-token tile)
__global__ __launch_bounds__(32)
void lora_shrink(const float* __restrict__ x, const float* __restrict__ lora_a,
                 const int* __restrict__ counts, const int* __restrict__ offsets,
                 const int* __restrict__ perm, float* __restrict__ vbuf) {
    const int a    = blockIdx.y;
    const int tile = blockIdx.x * 16;
    const int cnt  = counts[a];
    if (tile >= cnt) return;                 // wave-uniform exit
    const int seg  = offsets[a];
    const int lane = threadIdx.x;
    const int n    = lane & 15;              // A-frag: M = lane%16 ; D-frag: N = lane%16
    const int kh   = lane >> 4;              // K-half select (K = k0 + 2*kh + {0,1})
    const bool full = (tile + 16) <= cnt;    // wave-uniform: all 16 rows valid?

    // gather row for this lane's A-operand row M=n (clamped only on the tail tile)
    int mpos = tile + n;
    int arow = perm[seg + (full ? mpos : (mpos < cnt ? mpos : 0))];
    const float* xrow = x + (size_t)arow * H_DIM;
    const float* apan = lora_a + (size_t)a * (H_DIM * R_RANK);

    v8f acc = {0.f, 0.f, 0.f, 0.f, 0.f, 0.f, 0.f, 0.f};
    #pragma unroll 4
    for (int k0 = 0; k0 < H_DIM; k0 += 4) {
        const int kk = k0 + 2 * kh;
        v2f av = *(const v2f*)(xrow + kk);                    // A: 16x4 f32 frag
        v2f bv;                                               // B: 4x16 f32 frag
        bv.x = apan[(size_t)kk       * R_RANK + n];
        bv.y = apan[(size_t)(kk + 1) * R_RANK + n];
        acc = __builtin_amdgcn_wmma_f32_16x16x4_f32(
                  false, av, false, bv, (short)0, acc, false, false);
    }

    // D layout: VGPR v -> row M = v + 8*kh, col N = n.
    float* vb = vbuf + (size_t)(seg + tile + 8 * kh) * R_RANK + n;
    if (full) {                               // fast path: no predication
        #pragma unroll
        for (int v = 0; v < 8; ++v)
            vb[(size_t)v * R_RANK] = acc[v];
    } else {                                  // tail tile only
        #pragma unroll
        for (int v = 0; v < 8; ++v)
            if (tile + v + 8 * kh < cnt)
                vb[(size_t)v * R_RANK] = acc[v];
    }
}

// ---- expand helper: one wave handles 16-col strips of O for one 16-token tile
template <bool FULL>
__device__ __forceinline__
void expand_body(const float* __restrict__ resin, const float* __restrict__ bpan,
                 const int* __restrict__ perm, const float* __restrict__ vbuf,
                 float* __restrict__ out, int seg, int tile, int cnt,
                 int lane, int wave) {
    const int n  = lane & 15;
    const int kh = lane >> 4;

    // A-operand (V tile, K=16): hoisted, reused for every 16-column chunk.
    // Padding rows may read into the next segment's V — harmless, those D rows are dropped.
    const size_t vrow = (size_t)(seg + tile + n) * R_RANK;
    v2f afr[4];
    #pragma unroll
    for (int t4 = 0; t4 < 4; ++t4)
        afr[t4] = *(const v2f*)(vbuf + vrow + t4 * 4 + 2 * kh);

    // C/D row ids: VGPR v -> row M = v + 8*kh
    int rowIdx[8];
    #pragma unroll
    for (int v = 0; v < 8; ++v) {
        int m = v + 8 * kh;
        rowIdx[v] = perm[seg + (FULL ? (tile + m)
                                     : ((tile + m) < cnt ? (tile + m) : 0))];
    }

    for (int oc = wave * 16; oc < O_DIM; oc += 16 * 8) {
        v8f c;
        #pragma unroll
        for (int v = 0; v < 8; ++v)
            c[v] = resin[(size_t)rowIdx[v] * O_DIM + oc + n];   // C = base result rows

        #pragma unroll
        for (int t4 = 0; t4 < 4; ++t4) {                        // K=16 = 4 x wmma(K=4)
            const int kk = t4 * 4 + 2 * kh;
            v2f bv;
            bv.x = bpan[(size_t)kk       * O_DIM + oc + n];
            bv.y = bpan[(size_t)(kk + 1) * O_DIM + oc + n];
            c = __builtin_amdgcn_wmma_f32_16x16x4_f32(
                    false, afr[t4], false, bv, (short)0, c, false, false);
        }

        if (FULL) {                           // fast path: no exec-mask juggling
            #pragma unroll
            for (int v = 0; v < 8; ++v)
                out[(size_t)rowIdx[v] * O_DIM + oc + n] = c[v];
        } else {                              // at most one tail tile per segment
            #pragma unroll
            for (int v = 0; v < 8; ++v)
                if (tile + v + 8 * kh < cnt)
                    out[(size_t)rowIdx[v] * O_DIM + oc + n] = c[v];
        }
    }
}

// ---- expand: out[row] = result[row] + V[pos] @ lora_b[a]  (8 waves/tile, 16-col strips)
__global__ __launch_bounds__(256)
void lora_expand(const float* __restrict__ resin, const float* __restrict__ lora_b,
                 const int* __restrict__ counts, const int* __restrict__ offsets,
                 const int* __restrict__ perm, const float* __restrict__ vbuf,
                 float* __restrict__ out) {
    const int a    = blockIdx.y;
    const int tile = blockIdx.x * 16;
    const int cnt  = counts[a];
    if (tile >= cnt) return;                 // block-uniform exit
    const int seg  = offsets[a];
    const int lane = threadIdx.x & 31;
    const int wave = threadIdx.x >> 5;
    const float* bpan = lora_b + (size_t)a * (R_RANK * O_DIM);

    if ((tile + 16) <= cnt)                  // block-uniform branch
        expand_body<true >(resin, bpan, perm, vbuf, out, seg, tile, cnt, lane, wave);
    else
        expand_body<false>(resin, bpan, perm, vbuf, out, seg, tile, cnt, lane, wave);
}

extern "C" void kernel_launch(void* const* d_in, const int* in_sizes, int n_in,
                              void* d_out, int out_size, void* d_ws, size_t ws_size,
                              hipStream_t stream) {
    const float* resin  = (const float*)d_in[0];
    const float* x      = (const float*)d_in[1];
    const float* lora_a = (const float*)d_in[2];
    const float* lora_b = (const float*)d_in[3];
    const int*   idx    = (const int*)d_in[4];
    float*       out    = (float*)d_out;

    char*  ws      = (char*)d_ws;
    int*   counts  = (int*)(ws + WS_OFF_COUNTS);
    int*   offsets = (int*)(ws + WS_OFF_OFFSETS);
    int*   cursor  = (int*)(ws + WS_OFF_CURSOR);
    int*   perm    = (int*)(ws + WS_OFF_PERM);
    float* vbuf    = (float*)(ws + WS_OFF_VBUF);

    const int tblocks = (T_TOK + 255) / 256;
    const int ntiles  = (T_TOK + 15) / 16;   // worst case: all tokens on one adapter

    lora_init   <<<1, 32, 0, stream>>>(counts, cursor);
    lora_count  <<<tblocks, 256, 0, stream>>>(idx, counts);
    lora_prefix <<<1, 32, 0, stream>>>(counts, offsets);
    lora_scatter<<<tblocks, 256, 0, stream>>>(idx, offsets, cursor, perm);

    lora_shrink <<<dim3(ntiles, N_AD), 32, 0, stream>>>(x, lora_a, counts, offsets, perm, vbuf);
    lora_expand <<<dim3(ntiles, N_AD), 256, 0, stream>>>(resin, lora_b, counts, offsets, perm,
                                                         vbuf, out);
}